// GATFeatureExtractor_55353538511223
// MI455X (gfx1250) — compile-verified
//
#include <hip/hip_runtime.h>
#include <math.h>

// ---------------- problem constants (from reference) ----------------
#define NN      100000            // nodes
#define EE      1600000           // raw edges
#define ETOT    (EE + NN)         // edges + self loops
#define INC     128               // in channels
#define HIDC    32                // hidden channels per head
#define NHEADS  8
#define NEGSL   0.2f

typedef __attribute__((ext_vector_type(2))) float v2f;
typedef __attribute__((ext_vector_type(8))) float v8f;

// ---- ordered-key encoding so unsigned atomicMax == float max (all finite f) ----
__device__ __forceinline__ unsigned fkey(float f) {
    unsigned u = __float_as_uint(f);
    return (u & 0x80000000u) ? ~u : (u | 0x80000000u);
}
__device__ __forceinline__ float fdec(unsigned k) {
    unsigned u = (k & 0x80000000u) ? (k ^ 0x80000000u) : ~k;
    return __uint_as_float(u);
}
#define KEY_NEG_INF 0x007FFFFFu   // fkey(-inf)

__device__ __forceinline__ float lrelu(float x) { return x > 0.f ? x : NEGSL * x; }

// =====================================================================
// f32 GEMM via V_WMMA_F32_16X16X4_F32 (exact f32 math).
// grid.x = M/16 row tiles; blockDim.x = 32 * (Ncols/16); wave w -> col tile w.
// A:[M,K] row-major, B:[K,Ncols] row-major, C:[M,Ncols].
// =====================================================================
__global__ void wmma_gemm_f32(const float* __restrict__ A,
                              const float* __restrict__ B,
                              float* __restrict__ C,
                              int K, int Ncols) {
    const int wave = threadIdx.x >> 5;
    const int lane = threadIdx.x & 31;
    const int m    = lane & 15;          // row (A) or col (B/D) within tile
    const int half = lane >> 4;          // K-pair selector
    const size_t row0 = (size_t)blockIdx.x * 16;
    const int col0 = wave * 16;

    const float* arow = A + (row0 + (size_t)m) * (size_t)K;
    v8f acc = {};
    for (int k0 = 0; k0 < K; k0 += 4) {
        const int ka = k0 + 2 * half;
        v2f a, b;
        a.x = arow[ka];
        a.y = arow[ka + 1];
        b.x = B[(size_t)ka * Ncols + col0 + m];
        b.y = B[(size_t)(ka + 1) * Ncols + col0 + m];
        acc = __builtin_amdgcn_wmma_f32_16x16x4_f32(
            /*neg_a=*/false, a, /*neg_b=*/false, b,
            /*c_mod=*/(short)0, acc, /*reuse_a=*/false, /*reuse_b=*/false);
    }
    // D layout: lane L vgpr r -> (M = r + 8*half, N = m)
    float* crow = C + (row0 + 8 * (size_t)half) * (size_t)Ncols + col0 + m;
#pragma unroll
    for (int r = 0; r < 8; ++r) crow[(size_t)r * Ncols] = acc[r];
}

// =====================================================================
// per-(node,head) attention scores: e_src = xh[n,h,:].a_src[h,:], same for dst
// =====================================================================
__global__ void node_scores(const float* __restrict__ xh,
                            const float* __restrict__ a_src,
                            const float* __restrict__ a_dst,
                            float* __restrict__ es, float* __restrict__ ed,
                            int heads) {
    long long t = (long long)blockIdx.x * blockDim.x + threadIdx.x;  // n*heads+h
    if (t >= (long long)NN * heads) return;
    const int h = (int)(t % heads);
    const float* v = xh + (size_t)t * HIDC;
    float s = 0.f, d = 0.f;
#pragma unroll 8
    for (int c = 0; c < HIDC; ++c) {
        float xv = v[c];
        s += xv * a_src[h * HIDC + c];
        d += xv * a_dst[h * HIDC + c];
    }
    es[t] = s;
    ed[t] = d;
}

// =====================================================================
// init kernels (scratch must be re-initialized every call)
// =====================================================================
__global__ void init_keys_sums(unsigned* __restrict__ mkey, float* __restrict__ ssum,
                               long long n) {
    long long i = (long long)blockIdx.x * blockDim.x + threadIdx.x;
    if (i < n) { mkey[i] = KEY_NEG_INF; ssum[i] = 0.f; }
}
__global__ void init_bias_bcast(float* __restrict__ out, const float* __restrict__ bias,
                                int cmask, long long n) {
    long long i = (long long)blockIdx.x * blockDim.x + threadIdx.x;
    if (i < n) out[i] = bias[(int)(i & cmask)];
}

// =====================================================================
// edge pass 1: segment max of leaky-relu'd scores into dst (ordered-key atomicMax)
// =====================================================================
__global__ void edge_max(const int* __restrict__ ei,
                         const float* __restrict__ es, const float* __restrict__ ed,
                         unsigned* __restrict__ mkey, int heads) {
    long long t = (long long)blockIdx.x * blockDim.x + threadIdx.x;
    if (t >= (long long)ETOT) return;
    long long s, d;
    if (t < EE) { s = ei[t]; d = ei[EE + t]; } else { s = d = t - EE; }
    for (int h = 0; h < heads; ++h) {
        float e = lrelu(es[s * heads + h] + ed[d * heads + h]);
        atomicMax(&mkey[d * heads + h], fkey(e));
    }
}

// =====================================================================
// edge pass 2: segment sum of exp(e - max)
// =====================================================================
__global__ void edge_sum(const int* __restrict__ ei,
                         const float* __restrict__ es, const float* __restrict__ ed,
                         const unsigned* __restrict__ mkey, float* __restrict__ ssum,
                         int heads) {
    long long t = (long long)blockIdx.x * blockDim.x + threadIdx.x;
    if (t >= (long long)ETOT) return;
    long long s, d;
    if (t < EE) { s = ei[t]; d = ei[EE + t]; } else { s = d = t - EE; }
    for (int h = 0; h < heads; ++h) {
        float e = lrelu(es[s * heads + h] + ed[d * heads + h]);
        float p = expf(e - fdec(mkey[d * heads + h]));
        atomicAdd(&ssum[d * heads + h], p);
    }
}

// =====================================================================
// edge pass 3: out[dst,h,:] += alpha * xh[src,h,:]
// blockDim.x threads cover (blockDim.x / (heads*HIDC)) edges, one thread per channel.
// Accumulator is L2-resident (102MB / 12.8MB << 192MB L2).
// =====================================================================
__global__ void edge_agg(const int* __restrict__ ei,
                         const float* __restrict__ es, const float* __restrict__ ed,
                         const unsigned* __restrict__ mkey, const float* __restrict__ ssum,
                         const float* __restrict__ xh, float* __restrict__ out,
                         int heads) {
    const int HC  = heads * HIDC;
    const int epb = blockDim.x / HC;
    long long t = (long long)blockIdx.x * epb + threadIdx.x / HC;
    if (t >= (long long)ETOT) return;
    const int c = threadIdx.x % HC;
    const int h = c / HIDC;
    long long s, d;
    if (t < EE) { s = ei[t]; d = ei[EE + t]; } else { s = d = t - EE; }
    float e = lrelu(es[s * heads + h] + ed[d * heads + h]);
    float alpha = expf(e - fdec(mkey[d * heads + h])) / (ssum[d * heads + h] + 1e-16f);
    atomicAdd(&out[d * (long long)HC + c], xh[s * (long long)HC + c] * alpha);
}

__global__ void relu_k(float* __restrict__ x, long long n) {
    long long i = (long long)blockIdx.x * blockDim.x + threadIdx.x;
    if (i < n) x[i] = fmaxf(x[i], 0.f);
}

// =====================================================================
// host side
// =====================================================================
extern "C" void kernel_launch(void* const* d_in, const int* in_sizes, int n_in,
                              void* d_out, int out_size, void* d_ws, size_t ws_size,
                              hipStream_t stream) {
    (void)in_sizes; (void)n_in; (void)out_size; (void)ws_size;
    const float* x   = (const float*)d_in[0];
    const int*   ei  = (const int*)d_in[1];    // [2,E] flat: row0=src, row1=dst
    const float* W1  = (const float*)d_in[2];
    const float* as1 = (const float*)d_in[3];
    const float* ad1 = (const float*)d_in[4];
    const float* b1  = (const float*)d_in[5];
    const float* W2  = (const float*)d_in[6];
    const float* as2 = (const float*)d_in[7];
    const float* ad2 = (const float*)d_in[8];
    const float* b2  = (const float*)d_in[9];
    float* out = (float*)d_out;                // [NN, 32] flat (== reshape view)

    // ---- workspace layout (floats); ~218 MB total ----
    float* ws = (float*)d_ws;
    size_t o = 0;
    float* xh1  = ws + o; o += (size_t)NN * 256;   // layer1 xh  [N,8,32]
    float* hbuf = ws + o; o += (size_t)NN * 256;   // layer1 out / relu'd h
    float* es1  = ws + o; o += (size_t)NN * 8;
    float* ed1  = ws + o; o += (size_t)NN * 8;
    unsigned* m1 = (unsigned*)(ws + o); o += (size_t)NN * 8;
    float* s1   = ws + o; o += (size_t)NN * 8;
    // layer 2 reuses xh1's block (dead after edge_agg layer 1)
    float* xh2  = xh1;                              // [N,32]
    float* es2  = xh1 + (size_t)NN * 32;            // [N]
    float* ed2  = es2 + NN;                         // [N]
    unsigned* m2 = (unsigned*)(ed2 + NN);           // [N]
    float* s2   = ed2 + 2 * (size_t)NN;             // [N]

    const int TB = 256;
    const long long NH1 = (long long)NN * NHEADS;

    // -------- layer 1 --------
    wmma_gemm_f32<<<NN / 16, 16 * 32, 0, stream>>>(x, W1, xh1, INC, 256);
    node_scores<<<(unsigned)((NH1 + TB - 1) / TB), TB, 0, stream>>>(xh1, as1, ad1, es1, ed1, NHEADS);
    init_keys_sums<<<(unsigned)((NH1 + TB - 1) / TB), TB, 0, stream>>>(m1, s1, NH1);
    init_bias_bcast<<<(unsigned)(((long long)NN * 256 + TB - 1) / TB), TB, 0, stream>>>(hbuf, b1, 255, (long long)NN * 256);
    edge_max<<<(ETOT + TB - 1) / TB, TB, 0, stream>>>(ei, es1, ed1, m1, NHEADS);
    edge_sum<<<(ETOT + TB - 1) / TB, TB, 0, stream>>>(ei, es1, ed1, m1, s1, NHEADS);
    edge_agg<<<ETOT, 256, 0, stream>>>(ei, es1, ed1, m1, s1, xh1, hbuf, NHEADS);   // 1 edge/block
    relu_k<<<(unsigned)(((long long)NN * 256 + TB - 1) / TB), TB, 0, stream>>>(hbuf, (long long)NN * 256);

    // -------- layer 2 --------
    wmma_gemm_f32<<<NN / 16, 2 * 32, 0, stream>>>(hbuf, W2, xh2, 256, 32);
    node_scores<<<(NN + TB - 1) / TB, TB, 0, stream>>>(xh2, as2, ad2, es2, ed2, 1);
    init_keys_sums<<<(NN + TB - 1) / TB, TB, 0, stream>>>(m2, s2, (long long)NN);
    init_bias_bcast<<<(unsigned)(((long long)NN * 32 + TB - 1) / TB), TB, 0, stream>>>(out, b2, 31, (long long)NN * 32);
    edge_max<<<(ETOT + TB - 1) / TB, TB, 0, stream>>>(ei, es2, ed2, m2, 1);
    edge_sum<<<(ETOT + TB - 1) / TB, TB, 0, stream>>>(ei, es2, ed2, m2, s2, 1);
    edge_agg<<<(ETOT + 7) / 8, 256, 0, stream>>>(ei, es2, ed2, m2, s2, xh2, out, 1); // 8 edges/block
}